// TwolinksArmDynamic_67508295958658
// MI455X (gfx1250) — compile-verified
//
#include <hip/hip_runtime.h>
#include <hip/hip_bf16.h>
#include <stdint.h>

// ---------------- problem constants (match reference) ----------------
#define HORIZON 100000
#define DT      (1.0f / (float)HORIZON)

#define K1c 0.035f
#define K2c 0.045f
#define L1c 0.3f
#define M2c 1.0f
#define D2c 0.16f
#define A1c (K1c + K2c + M2c * L1c * L1c)   // 0.17
#define A2c (M2c * L1c * D2c)               // 0.048
#define A3c (K2c)                           // 0.045
#define B00 0.05f
#define B01 0.025f
#define B10 0.025f
#define B11 0.05f

// ---------------- streaming configuration ----------------
// CHUNK steps of (u0,u1) pairs per LDS buffer: 1024 steps * 8 B = 8 KB.
// Double buffered -> 16 KB LDS (WGP has 320 KB; trivial).
// One wave32 issues CHUNK/32 = 32 async b64 ops per chunk (8 B per lane).
#define CHUNK   1024
#define NCHUNK  ((HORIZON + CHUNK - 1) / CHUNK)   // 98 (97 full + 672-step tail)

// Issue one chunk of async global->LDS copies. GVS addressing:
//   global addr = SGPR64(controls) + VGPR32(byte offset)
//   LDS   addr = VGPR32 (byte offset within LDS)
// Tracked by ASYNCcnt; lanes past HORIZON are EXEC-masked off by the guard.
__device__ __forceinline__ void async_copy_chunk(uint64_t sbase, uint32_t lds_base,
                                                 int chunk_start, int lane) {
#pragma unroll
    for (int j = 0; j < CHUNK / 32; ++j) {
        int step = chunk_start + j * 32 + lane;
        if (step < HORIZON) {
            uint32_t l = lds_base + (uint32_t)(j * 32 + lane) * 8u;
            uint32_t g = (uint32_t)step * 8u;
            asm volatile("global_load_async_to_lds_b64 %0, %1, %2"
                         :: "v"(l), "v"(g), "s"(sbase)
                         : "memory");
        }
    }
}

__device__ __forceinline__ void wait_async_zero() {
    asm volatile("s_wait_asynccnt 0x0" ::: "memory");
}

__global__ __launch_bounds__(32)
void TwolinksArmDynamic_scan_kernel(const float* __restrict__ init_state,
                                    const float* __restrict__ controls,
                                    float* __restrict__ out) {
    __shared__ float smem[2][CHUNK * 2];
    const int lane = threadIdx.x;

    // State replicated across all 32 lanes (serial chain; redundant compute
    // keeps EXEC full and avoids divergence around the async issue slots).
    float x0 = init_state[0];
    float x1 = init_state[1];
    float x2 = init_state[2];
    float x3 = init_state[3];

    const uint64_t sbase = (uint64_t)(uintptr_t)controls;
    // LDS byte offsets of the two buffers (generic->LDS: low 32 bits = LDS addr).
    const uint32_t lds0 = (uint32_t)(uintptr_t)&smem[0][0];
    const uint32_t lds1 = (uint32_t)(uintptr_t)&smem[1][0];

    // Prefetch chunk 0 into buffer 0.
    async_copy_chunk(sbase, lds0, 0, lane);

    for (int c = 0; c < NCHUNK; ++c) {
        // All outstanding async ops belong to chunk c -> wait for it to land.
        wait_async_zero();
        // Kick off chunk c+1 into the other buffer; it streams from HBM/L2
        // while we burn through chunk c's 1024 dependent steps.
        if (c + 1 < NCHUNK)
            async_copy_chunk(sbase, (c & 1) ? lds0 : lds1, (c + 1) * CHUNK, lane);

        const float* __restrict__ buf = (c & 1) ? smem[1] : smem[0];
        int steps = HORIZON - c * CHUNK;
        if (steps > CHUNK) steps = CHUNK;

#pragma unroll 4
        for (int k = 0; k < steps; ++k) {
            const float u0 = buf[2 * k + 0];   // same addr all lanes -> LDS broadcast
            const float u1 = buf[2 * k + 1];

            // Native transcendentals: v_cos_f32 / v_sin_f32 (latency-critical).
            const float c2 = __cosf(x1);
            const float s2 = __sinf(x1);

            const float m12  = -(A3c + A2c * c2);
            const float m22  = A1c + 2.0f * A2c * c2;
            const float det  = A3c * (A1c - A3c) - (A2c * A2c) * (c2 * c2);
            const float rdet = __builtin_amdgcn_rcpf(det);   // v_rcp_f32; det>0 always

            const float C0  = -x3 * (2.0f * x2 + x3);
            const float C1  = x2 * x2;
            const float a2s = A2c * s2;

            const float aux0 = u0 - a2s * C0 - (B00 * x2 + B01 * x3);
            const float aux1 = u1 - a2s * C1 - (B10 * x2 + B11 * x3);

            const float nx0 = __builtin_fmaf(DT, x2, x0);
            const float nx1 = __builtin_fmaf(DT, x3, x1);
            const float dd  = DT * rdet;
            const float nx2 = __builtin_fmaf(dd, __builtin_fmaf(A3c, aux0, m12 * aux1), x2);
            const float nx3 = __builtin_fmaf(dd, __builtin_fmaf(m12, aux0, m22 * aux1), x3);

            x0 = nx0; x1 = nx1; x2 = nx2; x3 = nx3;
        }
    }

    if (lane == 0) {
        out[0] = x0;
        out[1] = x1;
        out[2] = x2;
        out[3] = x3;
    }
}

extern "C" void kernel_launch(void* const* d_in, const int* in_sizes, int n_in,
                              void* d_out, int out_size, void* d_ws, size_t ws_size,
                              hipStream_t stream) {
    (void)in_sizes; (void)n_in; (void)out_size; (void)d_ws; (void)ws_size;
    const float* init_state = (const float*)d_in[0];   // 4 floats
    const float* controls   = (const float*)d_in[1];   // HORIZON*2 floats
    float* out = (float*)d_out;                        // 4 floats

    // Strictly sequential recurrence: one wave32 owns the whole chain.
    TwolinksArmDynamic_scan_kernel<<<1, 32, 0, stream>>>(init_state, controls, out);
}